// TopKClassifer_73830487818735
// MI455X (gfx1250) — compile-verified
//
#include <hip/hip_runtime.h>
#include <hip/hip_bf16.h>
#include <math.h>

typedef __attribute__((ext_vector_type(2))) float v2f;
typedef __attribute__((ext_vector_type(8))) float v8f;
typedef unsigned int u32x4 __attribute__((ext_vector_type(4)));
typedef int i32x4 __attribute__((ext_vector_type(4)));
typedef int i32x8 __attribute__((ext_vector_type(8)));

#define BGR   32      // graphs
#define NPER0 2048    // nodes per graph, layer-0 input
#define HDIM  128
#define CIN   64
#define EDGES 1048576

// ---------------------------------------------------------------------------
// TDM: async 2D tile load (rows x 16 f32, row stride = strideElems) into LDS.
// ---------------------------------------------------------------------------
#if defined(__gfx1250__) && __has_builtin(__builtin_amdgcn_tensor_load_to_lds) && \
    __has_builtin(__builtin_amdgcn_s_wait_tensorcnt)
#define HAVE_TDM 1
__device__ __forceinline__ void tdm_load_tile16(void* ldsPtr, const float* gsrc,
                                                int rows, int strideElems) {
  const unsigned long long ga = (unsigned long long)(uintptr_t)gsrc;
  const unsigned int lds = (unsigned int)(uintptr_t)ldsPtr;  // low 32b == LDS byte offset
  u32x4 g0;
  g0[0] = 1u;                                                 // count=1 (valid user D#)
  g0[1] = lds;                                                // lds_addr
  g0[2] = (unsigned int)ga;                                   // global_addr[31:0]
  g0[3] = (unsigned int)((ga >> 32) & 0x01FFFFFFu) | (2u << 30);  // addr[56:32] | type=2
  i32x8 g1;
  g1[0] = (int)(2u << 16);                  // data_size = 4 bytes
  g1[1] = (int)(16u << 16);                 // tensor_dim0 = 16      (bits 79:48 lo)
  g1[2] = (int)((unsigned int)rows << 16);  // tensor_dim1 = rows    (bits 111:80 lo)
  g1[3] = (int)(16u << 16);                 // tile_dim0 = 16        (bits 127:112)
  g1[4] = (int)((unsigned int)rows & 0xFFFFu);  // tile_dim1 = rows  (bits 143:128)
  g1[5] = (int)(unsigned int)strideElems;   // tensor_dim0_stride lo32
  g1[6] = 0;
  g1[7] = 0;
  i32x4 gz; gz[0] = gz[1] = gz[2] = gz[3] = 0;
#if __clang_major__ >= 23
  i32x8 gz8; gz8[0]=gz8[1]=gz8[2]=gz8[3]=gz8[4]=gz8[5]=gz8[6]=gz8[7]=0;
  __builtin_amdgcn_tensor_load_to_lds(g0, g1, gz, gz, gz8, 0);
#else
  __builtin_amdgcn_tensor_load_to_lds(g0, g1, gz, gz, 0);
#endif
}
#endif

// ---------------------------------------------------------------------------
// fp32 WMMA GEMM: C[M,N] = relu?( A[M,K]@W[K,N] (+ A2[M,K2]@W2[K2,N]) + bias )
// Block = 8 waves; each wave owns one 16x16 tile, block covers 128 rows x 16 cols.
// W tiles staged in LDS via Tensor Data Mover. N multiple of 16 (no guards),
// M multiple of 16, K/K2 multiples of 4, K<=256, K2<=128.
// ---------------------------------------------------------------------------
__global__ __launch_bounds__(256) void wmma_gemm_f32(
    const float* __restrict__ A, const float* __restrict__ W,
    const float* __restrict__ A2, const float* __restrict__ W2,
    const float* __restrict__ bias, float* __restrict__ C,
    int M, int K, int K2, int N, int relu) {
  __shared__ float sW[256 * 16];
  __shared__ float sW2[128 * 16];
  const int n0 = blockIdx.x * 16;
  const int wave = threadIdx.x >> 5;
  const int lane = threadIdx.x & 31;
  const int m0 = blockIdx.y * 128 + wave * 16;
  const int r = lane & 15;   // row within A tile; col within B/C tile
  const int hi = lane >> 4;  // K-half select

#if defined(HAVE_TDM)
  if (threadIdx.x < 32) {  // one wave issues the DMA
    tdm_load_tile16(sW, W + n0, K, N);
    if (A2) tdm_load_tile16(sW2, W2 + n0, K2, N);
    __builtin_amdgcn_s_wait_tensorcnt(0);
  }
#else
  for (int idx = threadIdx.x; idx < K * 16; idx += 256)
    sW[idx] = W[(size_t)(idx >> 4) * N + n0 + (idx & 15)];
  if (A2)
    for (int idx = threadIdx.x; idx < K2 * 16; idx += 256)
      sW2[idx] = W2[(size_t)(idx >> 4) * N + n0 + (idx & 15)];
#endif
  __syncthreads();

  if (m0 >= M) return;  // wave-uniform: EXEC stays all-ones for WMMA

  v8f acc = {};
  {
    const float* arow = A + (size_t)(m0 + r) * K + 2 * hi;
#pragma unroll 4
    for (int kk = 0; kk < K; kk += 4) {
      v2f a, b;
      a.x = arow[kk];
      a.y = arow[kk + 1];
      b.x = sW[(kk + 2 * hi) * 16 + r];
      b.y = sW[(kk + 2 * hi + 1) * 16 + r];
      acc = __builtin_amdgcn_wmma_f32_16x16x4_f32(false, a, false, b,
                                                  (short)0, acc, false, false);
    }
  }
  if (A2) {
    const float* arow = A2 + (size_t)(m0 + r) * K2 + 2 * hi;
#pragma unroll 4
    for (int kk = 0; kk < K2; kk += 4) {
      v2f a, b;
      a.x = arow[kk];
      a.y = arow[kk + 1];
      b.x = sW2[(kk + 2 * hi) * 16 + r];
      b.y = sW2[(kk + 2 * hi + 1) * 16 + r];
      acc = __builtin_amdgcn_wmma_f32_16x16x4_f32(false, a, false, b,
                                                  (short)0, acc, false, false);
    }
  }
  const float bval = bias ? bias[n0 + r] : 0.0f;
#pragma unroll
  for (int i = 0; i < 8; ++i) {
    const int row = m0 + i + 8 * hi;  // C/D: VGPR i -> M=i (lanes 0-15) / M=i+8
    float v = acc[i] + bval;
    if (relu) v = fmaxf(v, 0.0f);
    C[(size_t)row * N + n0 + r] = v;
  }
}

// ---------------------------------------------------------------------------
__global__ void fill_f32(float* __restrict__ p, float v, int n) {
  int i = blockIdx.x * blockDim.x + threadIdx.x;
  if (i < n) p[i] = v;
}

// Edge scatter-add: agg[dst[e], c] += x[src[e], c] * mask[e]; Cin = 1<<cshift
__global__ void scatter_add(const float* __restrict__ x, const int* __restrict__ src,
                            const int* __restrict__ dst, const float* __restrict__ mask,
                            float* __restrict__ agg, int cshift, unsigned total) {
  unsigned i = blockIdx.x * blockDim.x + threadIdx.x;
  if (i >= total) return;
  const int e = (int)(i >> cshift);
  const int c = (int)(i & ((1u << cshift) - 1u));
  const float m = mask[e];
  if (m == 0.0f) return;
  atomicAdd(&agg[((size_t)dst[e] << cshift) + c],
            x[((size_t)src[e] << cshift) + c] * m);
}

// 1/||pool_w|| for the three layers (one wave)
__global__ void pw_invnorm(const float* __restrict__ p0, const float* __restrict__ p1,
                           const float* __restrict__ p2, float* __restrict__ invn) {
  const int lane = threadIdx.x;
  const float* ps[3] = {p0, p1, p2};
  for (int l = 0; l < 3; ++l) {
    float s = 0.0f;
    for (int j = lane; j < HDIM; j += 32) { float v = ps[l][j]; s += v * v; }
    for (int off = 16; off > 0; off >>= 1) s += __shfl_xor(s, off, 32);
    if (lane == 0) invn[l] = 1.0f / sqrtf(s);
  }
}

// score[v] = tanh(dot(y[v], pw) * invnorm); one wave32 per node
__global__ void score_kernel(const float* __restrict__ y, const float* __restrict__ pw,
                             const float* __restrict__ invn, int li,
                             float* __restrict__ score, int n) {
  const int wave = (int)((blockIdx.x * blockDim.x + threadIdx.x) >> 5);
  const int lane = threadIdx.x & 31;
  if (wave >= n) return;
  const float* row = y + (size_t)wave * HDIM;
  float s = row[lane]      * pw[lane]      + row[lane + 32] * pw[lane + 32] +
            row[lane + 64] * pw[lane + 64] + row[lane + 96] * pw[lane + 96];
  for (int off = 16; off > 0; off >>= 1) s += __shfl_xor(s, off, 32);
  if (lane == 0) score[wave] = tanhf(s * invn[li]);
}

// Per-graph descending bitonic sort in LDS; emit perm/vals/mapping
__global__ __launch_bounds__(256) void topk_kernel(
    const float* __restrict__ score, int n_per, int k,
    int* __restrict__ perm, float* __restrict__ vals, int* __restrict__ mapping) {
  __shared__ float skey[2048];
  __shared__ int sidx[2048];
  const int g = blockIdx.x;
  const int base = g * n_per;
  for (int j = threadIdx.x; j < n_per; j += blockDim.x) {
    skey[j] = score[base + j];
    sidx[j] = j;
    mapping[base + j] = -1;
  }
  __syncthreads();
  for (int size = 2; size <= n_per; size <<= 1) {
    for (int stride = size >> 1; stride > 0; stride >>= 1) {
      for (int i = threadIdx.x; i < (n_per >> 1); i += blockDim.x) {
        const int pos = 2 * i - (i & (stride - 1));
        const int par = pos + stride;
        const bool desc = ((pos & size) == 0);
        const float k0 = skey[pos], k1 = skey[par];
        if (desc ? (k0 < k1) : (k0 > k1)) {
          skey[pos] = k1; skey[par] = k0;
          const int t = sidx[pos]; sidx[pos] = sidx[par]; sidx[par] = t;
        }
      }
      __syncthreads();
    }
  }
  for (int j = threadIdx.x; j < k; j += blockDim.x) {
    const int oldLocal = sidx[j];
    perm[g * k + j] = base + oldLocal;
    vals[g * k + j] = skey[j];
    mapping[base + oldLocal] = g * k + j;
  }
}

// xn[i,c] = y[perm[i], c] * vals[i]
__global__ void gather_scale(const float* __restrict__ y, const int* __restrict__ perm,
                             const float* __restrict__ vals, float* __restrict__ xn,
                             unsigned total) {
  unsigned i = blockIdx.x * blockDim.x + threadIdx.x;
  if (i >= total) return;
  const int node = (int)(i >> 7), c = (int)(i & 127u);
  xn[i] = y[((size_t)perm[node]) * HDIM + c] * vals[node];
}

__global__ void relabel(const int* __restrict__ src, const int* __restrict__ dst,
                        const float* __restrict__ mask, const int* __restrict__ mapping,
                        int* __restrict__ nsrc, int* __restrict__ ndst,
                        float* __restrict__ nmask, int ne) {
  int e = blockIdx.x * blockDim.x + threadIdx.x;
  if (e >= ne) return;
  const int ns = mapping[src[e]], nd = mapping[dst[e]];
  const bool valid = (ns >= 0) && (nd >= 0);
  nmask[e] = valid ? mask[e] : 0.0f;
  nsrc[e] = valid ? ns : 0;
  ndst[e] = valid ? nd : 0;
}

// xs[g, 0:128] += max_j xn[g,j,:]; xs[g, 128:256] += mean_j xn[g,j,:]
__global__ __launch_bounds__(128) void readout(const float* __restrict__ xn,
                                               float* __restrict__ xs, int k) {
  const int g = blockIdx.x, c = threadIdx.x;
  const float* p = xn + (size_t)g * k * HDIM + c;
  float mx = -INFINITY, sm = 0.0f;
  for (int j = 0; j < k; ++j) {
    const float v = p[(size_t)j * HDIM];
    mx = fmaxf(mx, v);
    sm += v;
  }
  xs[g * 256 + c] += mx;
  xs[g * 256 + 128 + c] += sm / (float)k;
}

// Pad w3 [64,10]->[64,16] and b3 [10]->[16] with zeros so the last GEMM is guard-free
__global__ void pad_w3b3(const float* __restrict__ w3, const float* __restrict__ b3,
                         float* __restrict__ w3p, float* __restrict__ b3p) {
  int i = blockIdx.x * blockDim.x + threadIdx.x;
  if (i < 64 * 16) {
    const int rr = i >> 4, c = i & 15;
    w3p[i] = (c < 10) ? w3[rr * 10 + c] : 0.0f;
  } else if (i < 64 * 16 + 16) {
    const int c = i - 64 * 16;
    b3p[c] = (c < 10) ? b3[c] : 0.0f;
  }
}

// Extract [32,10] from padded [32,16]
__global__ void copy_out(const float* __restrict__ outp, float* __restrict__ out) {
  int i = threadIdx.x;  // 320 threads
  if (i < BGR * 10) out[i] = outp[(i / 10) * 16 + (i % 10)];
}

// ---------------------------------------------------------------------------
extern "C" void kernel_launch(void* const* d_in, const int* in_sizes, int n_in,
                              void* d_out, int out_size, void* d_ws, size_t ws_size,
                              hipStream_t stream) {
  const float* x0   = (const float*)d_in[0];
  const int*   src0 = (const int*)d_in[1];
  const int*   dst0 = (const int*)d_in[2];
  const float* msk0 = (const float*)d_in[3];
  const float* wrel[3]  = {(const float*)d_in[4], (const float*)d_in[5], (const float*)d_in[6]};
  const float* brel[3]  = {(const float*)d_in[7], (const float*)d_in[8], (const float*)d_in[9]};
  const float* wroot[3] = {(const float*)d_in[10], (const float*)d_in[11], (const float*)d_in[12]};
  const float* poolw[3] = {(const float*)d_in[13], (const float*)d_in[14], (const float*)d_in[15]};
  const float* w1 = (const float*)d_in[16]; const float* b1 = (const float*)d_in[17];
  const float* w2 = (const float*)d_in[18]; const float* b2 = (const float*)d_in[19];
  const float* w3 = (const float*)d_in[20]; const float* b3 = (const float*)d_in[21];
  float* out = (float*)d_out;

  // ---- workspace carve-out (256B aligned) ----
  char* ws = (char*)d_ws;
  size_t off = 0;
  auto carve = [&](size_t bytes) { void* p = ws + off; off += (bytes + 255) & ~(size_t)255; return p; };
  float* agg   = (float*)carve((size_t)65536 * 64 * 4);
  float* ybuf  = (float*)carve((size_t)65536 * HDIM * 4);
  float* xn    = (float*)carve((size_t)32768 * HDIM * 4);
  float* score = (float*)carve((size_t)65536 * 4);
  int*   perm  = (int*)carve((size_t)32768 * 4);
  float* vals  = (float*)carve((size_t)32768 * 4);
  int*   mapp  = (int*)carve((size_t)65536 * 4);
  int*   srcA  = (int*)carve((size_t)EDGES * 4);
  int*   dstA  = (int*)carve((size_t)EDGES * 4);
  float* mskA  = (float*)carve((size_t)EDGES * 4);
  int*   srcB  = (int*)carve((size_t)EDGES * 4);
  int*   dstB  = (int*)carve((size_t)EDGES * 4);
  float* mskB  = (float*)carve((size_t)EDGES * 4);
  float* xs    = (float*)carve((size_t)BGR * 256 * 4);
  float* h1    = (float*)carve((size_t)BGR * HDIM * 4);
  float* h2    = (float*)carve((size_t)BGR * 64 * 4);
  float* w3p   = (float*)carve((size_t)64 * 16 * 4);
  float* b3p   = (float*)carve((size_t)16 * 4);
  float* outp  = (float*)carve((size_t)BGR * 16 * 4);
  float* invn  = (float*)carve(256);
  (void)ws_size; (void)n_in; (void)in_sizes; (void)out_size;

  const int KS[3] = {1024, 512, 256};

  // init
  fill_f32<<<(BGR * 256 + 255) / 256, 256, 0, stream>>>(xs, 0.0f, BGR * 256);
  pw_invnorm<<<1, 32, 0, stream>>>(poolw[0], poolw[1], poolw[2], invn);
  pad_w3b3<<<5, 256, 0, stream>>>(w3, b3, w3p, b3p);

  const float* curX = x0;
  const int* curSrc = src0; const int* curDst = dst0; const float* curMsk = msk0;
  int n_per = NPER0;

  for (int li = 0; li < 3; ++li) {
    const int n = BGR * n_per;
    const int Cin = (li == 0) ? CIN : HDIM;
    const int cshift = (li == 0) ? 6 : 7;

    // agg = segment_sum(x[src]*mask, dst)
    fill_f32<<<((unsigned)n * Cin + 255) / 256, 256, 0, stream>>>(agg, 0.0f, n * Cin);
    {
      unsigned total = (unsigned)EDGES << cshift;
      scatter_add<<<(total + 255) / 256, 256, 0, stream>>>(curX, curSrc, curDst, curMsk,
                                                           agg, cshift, total);
    }

    // y = relu(agg@Wr + x@Wo + br)  (fused dual-GEMM, WMMA + TDM-staged weights)
    {
      dim3 grid(HDIM / 16, n / 128);
      wmma_gemm_f32<<<grid, 256, 0, stream>>>(agg, wrel[li], curX, wroot[li],
                                              brel[li], ybuf, n, Cin, Cin, HDIM, 1);
    }

    // scores + top-k per graph
    score_kernel<<<n / 8, 256, 0, stream>>>(ybuf, poolw[li], invn, li, score, n);
    const int k = KS[li];
    topk_kernel<<<BGR, 256, 0, stream>>>(score, n_per, k, perm, vals, mapp);

    // xn = y[perm] * vals
    {
      unsigned total = (unsigned)(BGR * k) * HDIM;
      gather_scale<<<(total + 255) / 256, 256, 0, stream>>>(ybuf, perm, vals, xn, total);
    }

    // relabel edges (not needed after the last layer)
    if (li == 0) {
      relabel<<<(EDGES + 255) / 256, 256, 0, stream>>>(curSrc, curDst, curMsk, mapp,
                                                       srcA, dstA, mskA, EDGES);
      curSrc = srcA; curDst = dstA; curMsk = mskA;
    } else if (li == 1) {
      relabel<<<(EDGES + 255) / 256, 256, 0, stream>>>(curSrc, curDst, curMsk, mapp,
                                                       srcB, dstB, mskB, EDGES);
      curSrc = srcB; curDst = dstB; curMsk = mskB;
    }

    // readout: xs += [max, mean]
    readout<<<BGR, HDIM, 0, stream>>>(xn, xs, k);

    curX = xn;
    n_per = k;
  }

  // MLP head: h1 = relu(xs@w1+b1); h2 = relu(h1@w2+b2); out = h2@w3+b3
  {
    dim3 g1(HDIM / 16, 1);
    wmma_gemm_f32<<<g1, 256, 0, stream>>>(xs, w1, nullptr, nullptr, b1, h1,
                                          BGR, 2 * HDIM, 0, HDIM, 1);
    dim3 g2(64 / 16, 1);
    wmma_gemm_f32<<<g2, 256, 0, stream>>>(h1, w2, nullptr, nullptr, b2, h2,
                                          BGR, HDIM, 0, 64, 1);
    dim3 g3(1, 1);  // padded N=16 tile, guard-free
    wmma_gemm_f32<<<g3, 256, 0, stream>>>(h2, w3p, nullptr, nullptr, b3p, outp,
                                          BGR, 64, 0, 16, 0);
    copy_out<<<1, 320, 0, stream>>>(outp, out);
  }
}